// MermaidNet_28741921145006
// MI455X (gfx1250) — compile-verified
//
#include <hip/hip_runtime.h>
#include <hip/hip_bf16.h>

typedef __attribute__((ext_vector_type(2))) float v2f;
typedef __attribute__((ext_vector_type(8))) float v8f;

// B element fetch, specialized at compile time (no runtime branching in the hot loop).
template <int CHAN>
__device__ __forceinline__ float fetchB(const float* __restrict__ I0,
                                        const float* __restrict__ I1,
                                        const float* __restrict__ F, long idx) {
    if constexpr (CHAN < 0)       return F[idx];
    else if constexpr (CHAN == 0) return I0[idx];
    else if constexpr (CHAN == 1) return I1[idx];
    else if constexpr (CHAN == 2) { float a = I0[idx]; return a * a; }
    else if constexpr (CHAN == 3) { float a = I1[idx]; return a * a; }
    else                          return I0[idx] * I1[idx];
}

// Banded GEMM pass: Out[m,n] = sum_k W[m,k] * B[k,n],
// W[m,k] = 1 iff k = s*m + 2*j, j in [0,ksz).
// One wave computes a 16x64 output tile: four 16x16 WMMA tiles sharing one A
// fragment (band weights generated in registers -> zero A memory traffic).
// Only the nonzero K band of the row tile is iterated.
// Main loop: unclamped loads via a single running 64-bit K index (strength-
// reduced addressing, provably in-bounds); short clamped epilogue for the tail.
// All masking flows through the A weights -> EXEC stays all-ones, branchless body.
template <int CHAN, bool KUNIT>
__global__ void __launch_bounds__(32)
lncc_pass_wmma(const float* __restrict__ I0, const float* __restrict__ I1,
               const float* __restrict__ Bsrc, float* __restrict__ Out,
               int Kdim, int M, int N,
               long strideK, long strideN, long zStrideB,
               long outStrideM, long outStrideN, long zStrideOut,
               int ksz, int s)
{
    const int lane = threadIdx.x;       // wave32
    const int hf   = lane >> 4;         // K-slot half select
    const int lid  = lane & 15;
    const int m0   = blockIdx.x << 4;
    const int n0   = blockIdx.y << 6;   // four 16-wide N tiles per wave
    const long bOff = (long)blockIdx.z * zStrideB;
    const long oOff = (long)blockIdx.z * zStrideOut;

    // Clamped column addresses: invalid columns read valid memory; their
    // products land in matmul columns that are never stored.
    int  nCol[4];
    long nTerm[4];
#pragma unroll
    for (int t = 0; t < 4; ++t) {
        nCol[t]  = n0 + 16 * t + lid;
        nTerm[t] = bOff + (long)(nCol[t] < N ? nCol[t] : N - 1) * strideN;
    }

    const int mRow = m0 + lid;          // A row held by this lane

    v8f acc[4];
#pragma unroll
    for (int t = 0; t < 4; ++t) acc[t] = (v8f){0.f,0.f,0.f,0.f,0.f,0.f,0.f,0.f};

    const int kBeg = s * m0;
    int kEnd = s * (m0 + 15) + 2 * (ksz - 1) + 1;
    if (kEnd > Kdim) kEnd = Kdim;

    // Band weights. k advances by 4 per iteration, so per-slot parity of
    // t = k - s*m is loop-invariant; fold it (and the mRow<M guard) into
    // per-lane constant masks. Per iteration each slot needs only one
    // unsigned range compare: (unsigned)t < 2*ksz <=> (t>=0 && t<2*ksz).
    int tRun = (kBeg + 2 * hf) - s * mRow;
    const float maskX = ((mRow < M) && ((tRun & 1) == 0))       ? 1.0f : 0.0f;
    const float maskY = ((mRow < M) && (((tRun + 1) & 1) == 0)) ? 1.0f : 0.0f;
    const unsigned wspan = (unsigned)(2 * ksz);

    auto wpair = [&](int tr) -> v2f {
        v2f a;
        a.x = ((unsigned)tr       < wspan) ? maskX : 0.0f;
        a.y = ((unsigned)(tr + 1) < wspan) ? maskY : 0.0f;
        return a;
    };

    // Clamped (always-safe) B fragment load, used in prologue and epilogue only.
    auto loadPairClamped = [&](int k0, long nT) -> v2f {
        int ka = k0 + 2 * hf;
        v2f b;
        if constexpr (KUNIT) {
            // strideK == 1: contiguous pair; VALID conv => max in-window
            // k <= Kdim-2, so clamping the base only perturbs zero-weight slots.
            int kc = ka > Kdim - 2 ? Kdim - 2 : ka;
            const float* p = Bsrc + kc + nT;
            b.x = p[0];
            b.y = p[1];
        } else {
            int ka_c = ka     > Kdim - 1 ? Kdim - 1 : ka;
            int kb_c = ka + 1 > Kdim - 1 ? Kdim - 1 : ka + 1;
            b.x = fetchB<CHAN>(I0, I1, Bsrc, (long)ka_c * strideK + nT);
            b.y = fetchB<CHAN>(I0, I1, Bsrc, (long)kb_c * strideK + nT);
        }
        return b;
    };

    // Software-pipelined band loop.
    int k0 = kBeg;
    v2f bcur[4];
#pragma unroll
    for (int t = 0; t < 4; ++t) bcur[t] = loadPairClamped(k0, nTerm[t]);

    // Running slot-x element index for k0; advances by loop-invariant kStep.
    long kIdx = (long)(kBeg + 2 * hf) * strideK;
    const long kStep = 4 * strideK;

    // Main loop: prefetch slots (k0+4 .. k0+7) are in-bounds -> no clamping,
    // pure incremental addressing.
    for (; (k0 + 4 < kEnd) && (k0 + 8 <= Kdim); k0 += 4, tRun += 4) {
        const long p = kIdx + kStep;
        v2f bnext[4];
#pragma unroll
        for (int t = 0; t < 4; ++t) {
            if constexpr (KUNIT) {
                const float* q = Bsrc + p + nTerm[t];
                bnext[t].x = q[0];
                bnext[t].y = q[1];
            } else {
                bnext[t].x = fetchB<CHAN>(I0, I1, Bsrc, p + nTerm[t]);
                bnext[t].y = fetchB<CHAN>(I0, I1, Bsrc, p + strideK + nTerm[t]);
            }
        }
        v2f a = wpair(tRun);
#pragma unroll
        for (int t = 0; t < 4; ++t)
            acc[t] = __builtin_amdgcn_wmma_f32_16x16x4_f32(
                         false, a, false, bcur[t], (short)0, acc[t], false, false);
#pragma unroll
        for (int t = 0; t < 4; ++t) bcur[t] = bnext[t];
        kIdx = p;
    }
    // Epilogue: remaining (<=2) iterations with clamped prefetch.
    for (; k0 + 4 < kEnd; k0 += 4, tRun += 4) {
        v2f bnext[4];
#pragma unroll
        for (int t = 0; t < 4; ++t) bnext[t] = loadPairClamped(k0 + 4, nTerm[t]);
        v2f a = wpair(tRun);
#pragma unroll
        for (int t = 0; t < 4; ++t)
            acc[t] = __builtin_amdgcn_wmma_f32_16x16x4_f32(
                         false, a, false, bcur[t], (short)0, acc[t], false, false);
#pragma unroll
        for (int t = 0; t < 4; ++t) bcur[t] = bnext[t];
    }
    {
        v2f a = wpair(tRun);
#pragma unroll
        for (int t = 0; t < 4; ++t)
            acc[t] = __builtin_amdgcn_wmma_f32_16x16x4_f32(
                         false, a, false, bcur[t], (short)0, acc[t], false, false);
    }

    // D layout: VGPR r -> M = r (lanes 0-15) / r+8 (lanes 16-31); N = lane%16.
#pragma unroll
    for (int r = 0; r < 8; ++r) {
        int m = m0 + r + 8 * hf;
        if (m < M) {
            long rowOff = oOff + (long)m * outStrideM;
#pragma unroll
            for (int t = 0; t < 4; ++t)
                if (nCol[t] < N)
                    Out[rowOff + (long)nCol[t] * outStrideN] = acc[t][r];
        }
    }
}

// Stage 1: per-block partial LNCC sums (fixed grid -> deterministic).
__global__ void __launch_bounds__(256)
lncc_reduce(const float* __restrict__ T3, long Lc, float invNumel,
            float* __restrict__ partials)
{
    __shared__ float sm[256];
    float sum = 0.f;
    for (long i = (long)blockIdx.x * blockDim.x + threadIdx.x; i < Lc;
         i += (long)gridDim.x * blockDim.x) {
        float is = T3[i];
        float ts = T3[i + Lc];
        float i2 = T3[i + 2 * Lc];
        float t2 = T3[i + 3 * Lc];
        float it = T3[i + 4 * Lc];
        float cross = it - is * ts * invNumel;
        float ivar  = i2 - is * is * invNumel;
        float tvar  = t2 - ts * ts * invNumel;
        sum += cross * cross / (ivar * tvar + 1e-5f);
    }
    sm[threadIdx.x] = sum;
    __syncthreads();
    for (int off = 128; off > 0; off >>= 1) {
        if (threadIdx.x < off) sm[threadIdx.x] += sm[threadIdx.x + off];
        __syncthreads();
    }
    if (threadIdx.x == 0) partials[blockIdx.x] = sm[0];
}

// Stage 2: ordered deterministic finish; accumulate (1-mean)*w into d_out.
__global__ void lncc_final(const float* __restrict__ partials, int nPart,
                           float invCount, float w, int init,
                           float* __restrict__ out)
{
    if (threadIdx.x == 0 && blockIdx.x == 0) {
        float s = 0.f;
        for (int i = 0; i < nPart; ++i) s += partials[i];
        float contrib = (1.0f - s * invCount) * w;
        if (init) out[0] = contrib;
        else      out[0] += contrib;
    }
}

extern "C" void kernel_launch(void* const* d_in, const int* in_sizes, int n_in,
                              void* d_out, int out_size, void* d_ws, size_t ws_size,
                              hipStream_t stream) {
    const float* I0 = (const float*)d_in[0];
    const float* I1 = (const float*)d_in[1];
    float* out = (float*)d_out;
    char* ws = (char*)d_ws;

    // Workspace (bytes): T1c 8,404,992 | T2c 2,495,232 | T3 3,704,064 | partials 2KB
    float* T1 = (float*)(ws);
    float* T2 = (float*)(ws + 8404992);
    float* T3 = (float*)(ws + 8404992 + 2495232);
    float* partials = (float*)(ws + 8404992 + 2495232 + 3704064);

    const int   kk[3]  = {12, 24, 48};
    const int   ss[3]  = {3, 6, 12};
    const int   LL[3]  = {57, 25, 9};
    const float wgt[3] = {0.1f, 0.3f, 0.6f};
    const long  PLANE  = 192L * 192L;   // 36864

    for (int sc = 0; sc < 3; ++sc) {
        const int k = kk[sc], s = ss[sc], L = LL[sc];
        const long L3 = (long)L * L * L;
        const int mT = (L + 15) / 16;

        for (int c = 0; c < 5; ++c) {
            // Pass Z: reduce z from fused channel of I0/I1 -> T1[oz][y][x].
            dim3 gz((unsigned)mT, (unsigned)((PLANE + 63) / 64), 1);
            switch (c) {
            case 0: lncc_pass_wmma<0, false><<<gz, 32, 0, stream>>>(I0, I1, nullptr, T1,
                        192, L, (int)PLANE, PLANE, 1, 0, PLANE, 1, 0, k, s); break;
            case 1: lncc_pass_wmma<1, false><<<gz, 32, 0, stream>>>(I0, I1, nullptr, T1,
                        192, L, (int)PLANE, PLANE, 1, 0, PLANE, 1, 0, k, s); break;
            case 2: lncc_pass_wmma<2, false><<<gz, 32, 0, stream>>>(I0, I1, nullptr, T1,
                        192, L, (int)PLANE, PLANE, 1, 0, PLANE, 1, 0, k, s); break;
            case 3: lncc_pass_wmma<3, false><<<gz, 32, 0, stream>>>(I0, I1, nullptr, T1,
                        192, L, (int)PLANE, PLANE, 1, 0, PLANE, 1, 0, k, s); break;
            default: lncc_pass_wmma<4, false><<<gz, 32, 0, stream>>>(I0, I1, nullptr, T1,
                        192, L, (int)PLANE, PLANE, 1, 0, PLANE, 1, 0, k, s); break;
            }

            // Pass Y: reduce y per oz-plane. T1[oz][y][x] -> T2[oz][oy][x].
            dim3 gy((unsigned)mT, (192 + 63) / 64, (unsigned)L);
            lncc_pass_wmma<-1, false><<<gy, 32, 0, stream>>>(I0, I1, T1, T2,
                192, L, 192,
                /*strideK*/192, /*strideN*/1, /*zStrideB*/PLANE,
                /*outStrideM*/192, /*outStrideN*/1, /*zStrideOut*/(long)L * 192,
                k, s);

            // Pass X: reduce contiguous x. T2[oz][oy][x] -> T3[c][oz][oy][ox],
            // columns n = oz*L+oy; out addr = n*L + m.
            dim3 gx((unsigned)mT, (unsigned)((L * L + 63) / 64), 1);
            lncc_pass_wmma<-1, true><<<gx, 32, 0, stream>>>(I0, I1, T2, T3 + (long)c * L3,
                192, L, L * L,
                /*strideK*/1, /*strideN*/192, /*zStrideB*/0,
                /*outStrideM*/1, /*outStrideN*/L, /*zStrideOut*/0,
                k, s);
        }

        const float invNumel = 1.0f / ((float)k * (float)k * (float)k);
        lncc_reduce<<<512, 256, 0, stream>>>(T3, L3, invNumel, partials);
        lncc_final<<<1, 32, 0, stream>>>(partials, 512, 1.0f / (float)L3,
                                         wgt[sc], sc == 0 ? 1 : 0, out);
    }
}